// LevelSetMessageAggregator_69200513073318
// MI455X (gfx1250) — compile-verified
//
#include <hip/hip_runtime.h>
#include <hip/hip_bf16.h>
#include <math.h>

// Problem constants (from reference)
#define NUM_SRC   200000
#define NUM_DST   100000
#define NUM_EDGES 1600000
#define IN_DIM    128
#define HID       256
#define OUT_DIM   128

typedef float v2f __attribute__((ext_vector_type(2)));
typedef float v8f __attribute__((ext_vector_type(8)));

#define ROWS 64   // rows per workgroup (four 16-row WMMA tiles)
#define LDA  132  // padded LDS stride for 128-wide rows (conflict-free)
#define LDH  260  // padded LDS stride for 256-wide rows
#define NBLK ((NUM_DST + ROWS - 1) / ROWS)        // 1563 (last WG partial)
#define SMEM_BYTES ((ROWS * LDA + ROWS * LDH) * 4) // 100352 B dynamic LDS

// ---------------------------------------------------------------------------
// Kernel 0: zero the accumulation workspace (sum + cnt)
// ---------------------------------------------------------------------------
__global__ __launch_bounds__(256) void zero_kernel(float* __restrict__ p, size_t n) {
    size_t i = (size_t)blockIdx.x * blockDim.x + threadIdx.x;
    if (i < n) p[i] = 0.0f;
}

// ---------------------------------------------------------------------------
// Kernel 1: edge gather + scatter-add (segment sum) with f32 L2 atomics.
// One wave (32 lanes) per edge; each lane moves a float4 (128 dims / 32 = 4).
// ---------------------------------------------------------------------------
__global__ __launch_bounds__(256) void aggregate_kernel(
    const float* __restrict__ x,
    const int*   __restrict__ edge_src,
    const int*   __restrict__ edge_dst,
    float* __restrict__ sum,
    float* __restrict__ cnt,
    int num_edges)
{
    int edge = (int)((blockIdx.x * blockDim.x + threadIdx.x) >> 5);
    int lane = threadIdx.x & 31;
    if (edge >= num_edges) return;

    int src = edge_src[edge];   // uniform per wave -> scalar broadcast load
    int dst = edge_dst[edge];

    const float4 v = *(const float4*)(x + (size_t)src * IN_DIM + lane * 4);
    float* d = sum + (size_t)dst * IN_DIM + lane * 4;
    unsafeAtomicAdd(d + 0, v.x);
    unsafeAtomicAdd(d + 1, v.y);
    unsafeAtomicAdd(d + 2, v.z);
    unsafeAtomicAdd(d + 3, v.w);
    if (lane == 0) unsafeAtomicAdd(cnt + dst, 1.0f);
}

// ---------------------------------------------------------------------------
// Kernel 2: fused  mean -> (GEMM1 x2 + bias) -> LN1 -> GELU -> GEMM2 + bias
//           -> LN2 -> out.  64 rows per WG, 16 waves (512 threads).
// Each wave owns one 16-col stripe and FOUR 16-row tiles that share one
// B fragment per k-step (register reuse => 1/4 weight traffic per row).
// ---------------------------------------------------------------------------
__global__ __launch_bounds__(512) void fused_mlp_kernel(
    const float* __restrict__ sum, const float* __restrict__ cnt,
    const float* __restrict__ x,
    const float* __restrict__ w_l,  const float* __restrict__ b_l,
    const float* __restrict__ w_r,
    const float* __restrict__ ln1_g, const float* __restrict__ ln1_b,
    const float* __restrict__ w_out, const float* __restrict__ b_out,
    const float* __restrict__ ln2_g, const float* __restrict__ ln2_b,
    float* __restrict__ out)
{
    extern __shared__ float smem[];
    float* sA = smem;               // ROWS x LDA : mean, then x_tgt, then out
    float* sH = smem + ROWS * LDA;  // ROWS x LDH : hidden 64x256

    const int tid  = threadIdx.x;
    const int wave = tid >> 5;
    const int lane = tid & 31;
    const int rowbase = blockIdx.x * ROWS;

    const int n    = lane & 15;      // output column within 16-wide tile
    const int kh   = lane >> 4;      // K half-select (A/B fragment layout)
    const int arow = lane & 15;      // A row within 16-row tile

    // ---- Stage A = mean-neighbor tile (64 x 128): 4 float4 per thread ----
    #pragma unroll
    for (int j = 0; j < 4; ++j) {
        const int idx = tid + j * 512;
        const int row = idx >> 5;            // 0..63
        const int k   = (idx & 31) * 4;      // 0..124
        int rg = rowbase + row;              // clamp tail reads into valid range
        if (rg >= NUM_DST) rg = NUM_DST - 1;
        const float c   = cnt[rg];
        const float inv = 1.0f / fmaxf(c, 1.0f);
        const float4 s4 = *(const float4*)(sum + (size_t)rg * IN_DIM + k);
        sA[row * LDA + k + 0] = s4.x * inv;
        sA[row * LDA + k + 1] = s4.y * inv;
        sA[row * LDA + k + 2] = s4.z * inv;
        sA[row * LDA + k + 3] = s4.w * inv;
    }
    __syncthreads();

    v8f acc[4];                               // four 16-row tiles
    #pragma unroll
    for (int rt = 0; rt < 4; ++rt) acc[rt] = (v8f){};
    const int colbase = wave * 16;            // 16 waves x 16 cols = 256 = HID

    // ---- GEMM1 pass 1: mean @ w_l (one B fragment -> 4 WMMAs) ----
    #pragma unroll 2
    for (int k0 = 0; k0 < IN_DIM; k0 += 4) {
        const int kb = k0 + 2 * kh;
        v2f b;
        b.x = w_l[(size_t)kb * HID + colbase + n];
        b.y = w_l[(size_t)(kb + 1) * HID + colbase + n];
        #pragma unroll
        for (int rt = 0; rt < 4; ++rt) {
            v2f a;
            a.x = sA[(rt * 16 + arow) * LDA + kb];
            a.y = sA[(rt * 16 + arow) * LDA + kb + 1];
            acc[rt] = __builtin_amdgcn_wmma_f32_16x16x4_f32(
                false, a, false, b, (short)0, acc[rt], false, false);
        }
    }
    __syncthreads();

    // ---- Restage A = x_tgt tile (64 x 128) ----
    #pragma unroll
    for (int j = 0; j < 4; ++j) {
        const int idx = tid + j * 512;
        const int row = idx >> 5;
        const int k   = (idx & 31) * 4;
        int rg = rowbase + row;               // x has NUM_SRC >= NUM_DST rows:
        if (rg >= NUM_DST) rg = NUM_DST - 1;  // clamp anyway (values unused)
        const float4 x4 = *(const float4*)(x + (size_t)rg * IN_DIM + k);
        sA[row * LDA + k + 0] = x4.x;
        sA[row * LDA + k + 1] = x4.y;
        sA[row * LDA + k + 2] = x4.z;
        sA[row * LDA + k + 3] = x4.w;
    }
    __syncthreads();

    // ---- GEMM1 pass 2: + x_tgt @ w_r into the same accumulators ----
    #pragma unroll 2
    for (int k0 = 0; k0 < IN_DIM; k0 += 4) {
        const int kb = k0 + 2 * kh;
        v2f b;
        b.x = w_r[(size_t)kb * HID + colbase + n];
        b.y = w_r[(size_t)(kb + 1) * HID + colbase + n];
        #pragma unroll
        for (int rt = 0; rt < 4; ++rt) {
            v2f a;
            a.x = sA[(rt * 16 + arow) * LDA + kb];
            a.y = sA[(rt * 16 + arow) * LDA + kb + 1];
            acc[rt] = __builtin_amdgcn_wmma_f32_16x16x4_f32(
                false, a, false, b, (short)0, acc[rt], false, false);
        }
    }

    // ---- + bias, spill h tiles to LDS ----
    {
        const float bias = b_l[colbase + n];
        #pragma unroll
        for (int rt = 0; rt < 4; ++rt) {
            #pragma unroll
            for (int i = 0; i < 8; ++i) {
                const int row = rt * 16 + i + 8 * kh;  // C: VGPR i -> rows i, i+8
                sH[row * LDH + colbase + n] = acc[rt][i] + bias;
            }
        }
    }
    __syncthreads();

    // ---- LN1 + exact GELU: four rows per wave, 8 elements per lane ----
    #pragma unroll
    for (int rr = 0; rr < 4; ++rr) {
        const int row = wave + 16 * rr;       // 0..63
        float v[8], s = 0.0f, s2 = 0.0f;
        #pragma unroll
        for (int j = 0; j < 8; ++j) {
            const float t = sH[row * LDH + lane + 32 * j];
            v[j] = t; s += t; s2 += t * t;
        }
        #pragma unroll
        for (int off = 16; off > 0; off >>= 1) {
            s  += __shfl_xor(s,  off, 32);
            s2 += __shfl_xor(s2, off, 32);
        }
        const float mean = s * (1.0f / HID);
        const float var  = s2 * (1.0f / HID) - mean * mean;
        const float rstd = rsqrtf(var + 1e-5f);
        #pragma unroll
        for (int j = 0; j < 8; ++j) {
            const int col = lane + 32 * j;
            const float hn = (v[j] - mean) * rstd * ln1_g[col] + ln1_b[col];
            sH[row * LDH + col] = 0.5f * hn * (1.0f + erff(hn * 0.70710678118654752f));
        }
    }
    __syncthreads();

    // ---- GEMM2: out = h @ w_out + b_out (waves 0..7; B fragment reused
    //      across 4 row tiles; K = 256) ----
    float* sOut = sA;   // reuse A buffer (64 x 128, stride LDA)
    if (wave < 8) {
        const int cb2 = wave * 16;            // 8 waves x 16 cols = 128
        v8f c[4];
        #pragma unroll
        for (int rt = 0; rt < 4; ++rt) c[rt] = (v8f){};
        #pragma unroll 2
        for (int k0 = 0; k0 < HID; k0 += 4) {
            const int kb = k0 + 2 * kh;
            v2f b;
            b.x = w_out[(size_t)kb * OUT_DIM + cb2 + n];
            b.y = w_out[(size_t)(kb + 1) * OUT_DIM + cb2 + n];
            #pragma unroll
            for (int rt = 0; rt < 4; ++rt) {
                v2f a;
                a.x = sH[(rt * 16 + arow) * LDH + kb];
                a.y = sH[(rt * 16 + arow) * LDH + kb + 1];
                c[rt] = __builtin_amdgcn_wmma_f32_16x16x4_f32(
                    false, a, false, b, (short)0, c[rt], false, false);
            }
        }
        const float bias = b_out[cb2 + n];
        #pragma unroll
        for (int rt = 0; rt < 4; ++rt) {
            #pragma unroll
            for (int i = 0; i < 8; ++i) {
                const int row = rt * 16 + i + 8 * kh;
                sOut[row * LDA + cb2 + n] = c[rt][i] + bias;
            }
        }
    }
    __syncthreads();

    // ---- LN2 + store: four rows per wave, 4 elements per lane ----
    #pragma unroll
    for (int rr = 0; rr < 4; ++rr) {
        const int row = wave + 16 * rr;
        float v[4], s = 0.0f, s2 = 0.0f;
        #pragma unroll
        for (int j = 0; j < 4; ++j) {
            const float t = sOut[row * LDA + lane + 32 * j];
            v[j] = t; s += t; s2 += t * t;
        }
        #pragma unroll
        for (int off = 16; off > 0; off >>= 1) {
            s  += __shfl_xor(s,  off, 32);
            s2 += __shfl_xor(s2, off, 32);
        }
        const float mean = s * (1.0f / OUT_DIM);
        const float var  = s2 * (1.0f / OUT_DIM) - mean * mean;
        const float rstd = rsqrtf(var + 1e-5f);
        const int rowg = rowbase + row;
        if (rowg < NUM_DST) {
            #pragma unroll
            for (int j = 0; j < 4; ++j) {
                const int col = lane + 32 * j;
                const float o = (v[j] - mean) * rstd * ln2_g[col] + ln2_b[col];
                out[(size_t)rowg * OUT_DIM + col] = o;
            }
        }
    }
}

// ---------------------------------------------------------------------------
extern "C" void kernel_launch(void* const* d_in, const int* in_sizes, int n_in,
                              void* d_out, int out_size, void* d_ws, size_t ws_size,
                              hipStream_t stream) {
    const float* x      = (const float*)d_in[0];
    const float* w_l    = (const float*)d_in[1];
    const float* b_l    = (const float*)d_in[2];
    const float* w_r    = (const float*)d_in[3];
    const float* ln1_g  = (const float*)d_in[4];
    const float* ln1_b  = (const float*)d_in[5];
    const float* w_out  = (const float*)d_in[6];
    const float* b_out  = (const float*)d_in[7];
    const float* ln2_g  = (const float*)d_in[8];
    const float* ln2_b  = (const float*)d_in[9];
    const int*   e_src  = (const int*)d_in[10];
    const int*   e_dst  = (const int*)d_in[11];

    float* sum = (float*)d_ws;                       // NUM_DST x 128
    float* cnt = sum + (size_t)NUM_DST * IN_DIM;     // NUM_DST
    float* out = (float*)d_out;

    // 0) zero accumulators
    const size_t nzero = (size_t)NUM_DST * IN_DIM + NUM_DST;
    zero_kernel<<<(unsigned)((nzero + 255) / 256), 256, 0, stream>>>(sum, nzero);

    // 1) scatter-mean numerator/denominator: 1 wave per edge, 8 edges/block
    aggregate_kernel<<<NUM_EDGES / 8, 256, 0, stream>>>(x, e_src, e_dst, sum, cnt,
                                                        NUM_EDGES);

    // 2) fused MLP: 64 rows per WG, 1563 WGs, ~98 KB dynamic LDS
    fused_mlp_kernel<<<NBLK, 512, SMEM_BYTES, stream>>>(sum, cnt, x,
                                                        w_l, b_l, w_r,
                                                        ln1_g, ln1_b,
                                                        w_out, b_out,
                                                        ln2_g, ln2_b, out);
}